// GCP2_8546984919097
// MI455X (gfx1250) — compile-verified
//
#include <hip/hip_runtime.h>
#include <hip/hip_bf16.h>
#include <math.h>

typedef __attribute__((ext_vector_type(2))) float v2f;
typedef __attribute__((ext_vector_type(8))) float v8f;

#define S_IN 128
#define V_IN 16
#define S_OUT 128
#define V_OUT 16
#define HID 16
#define SVO 3
#define MERGED 153      // S_IN + HID + 3*SVO
#define MERGED_PAD 160
#define EPSC 1e-8f

// fast sigmoid: v_exp_f32 + v_rcp_f32 (no IEEE div sequence); ~1ulp rcp error,
// negligible vs fp32 reference tolerance
__device__ __forceinline__ float sigmoidf_(float x){
  return __builtin_amdgcn_rcpf(1.0f + __expf(-x));
}

// ---------------- kernel 0: zero scatter accumulators ----------------
__global__ void k_zero(float* p, int n){
  int i = blockIdx.x*blockDim.x + threadIdx.x;
  if (i < n) p[i] = 0.0f;
}

// ---------------- kernel 0b: pad W_so (128,153) -> (128,160) zero-filled ----------------
__global__ void k_pad_wso(const float* __restrict__ W_so, float* __restrict__ Wp){
  int i = blockIdx.x*blockDim.x + threadIdx.x;   // grid covers 128*160
  int row = i / MERGED_PAD, col = i % MERGED_PAD;
  Wp[i] = (col < MERGED) ? W_so[row*MERGED + col] : 0.0f;
}

// ---------------- kernel 1: per-node prep: vh, vnorm, vdf ----------------
// vh  (N,3,HID)  at n*48 + a*16 + h ; vnorm (N,16) ; vdf (N,3,SVO) at n*9 + a*3 + c
__global__ void k_node_prep(const float* __restrict__ vec,    // (N,16,3)
                            const float* __restrict__ W_down, // (16,16)
                            const float* __restrict__ W_df,   // (3,16)
                            float* __restrict__ vh,
                            float* __restrict__ vnorm,
                            float* __restrict__ vdf,
                            int N){
  __shared__ float vl[16*48];
  int nb = blockIdx.x*16;
  for (int idx = threadIdx.x; idx < 16*48; idx += 256){
    int n = nb + idx/48;
    vl[idx] = (n < N) ? vec[n*48 + idx%48] : 0.0f;
  }
  __syncthreads();
  int ln = threadIdx.x >> 4;   // local node 0..15
  int h  = threadIdx.x & 15;   // hidden idx 0..15
  int n = nb + ln;
  if (n >= N) return;
  const float* v = &vl[ln*48];
  float a0=0.f, a1=0.f, a2=0.f;
  #pragma unroll
  for (int i=0;i<16;i++){
    float w = W_down[h*16+i];
    a0 += v[i*3+0]*w; a1 += v[i*3+1]*w; a2 += v[i*3+2]*w;
  }
  vh[n*48 + 0*16 + h] = a0;
  vh[n*48 + 1*16 + h] = a1;
  vh[n*48 + 2*16 + h] = a2;
  vnorm[n*16+h] = sqrtf(a0*a0 + a1*a1 + a2*a2 + EPSC);
  if (h < SVO){
    float d0=0.f,d1=0.f,d2=0.f;
    #pragma unroll
    for (int i=0;i<16;i++){
      float w = W_df[h*16+i];
      d0 += v[i*3+0]*w; d1 += v[i*3+1]*w; d2 += v[i*3+2]*w;
    }
    vdf[n*9 + 0*3 + h] = d0;
    vdf[n*9 + 1*3 + h] = d1;
    vdf[n*9 + 2*3 + h] = d2;
  }
}

// ---------------- kernel 2: per-edge rotate + scatter-add (L2 atomics) ----------------
__global__ void k_edge_scatter(const int*   __restrict__ ei,     // (2,E), row = ei[e]
                               const float* __restrict__ frames, // (E,3,3)
                               const float* __restrict__ vdf,    // (N,9) [a*3+c]
                               float* __restrict__ sums,         // (N,9) [c*3+i]
                               float* __restrict__ cnt,          // (N)
                               int E){
  int e = blockIdx.x*blockDim.x + threadIdx.x;
  if (e >= E) return;
  int r = ei[e];
  float F[9], V[9];
  #pragma unroll
  for (int q=0;q<9;q++) F[q] = frames[(size_t)e*9+q];
  #pragma unroll
  for (int q=0;q<9;q++) V[q] = vdf[(size_t)r*9+q];
  #pragma unroll
  for (int i=0;i<3;i++){
    #pragma unroll
    for (int c=0;c<3;c++){
      float lv = F[i*3+0]*V[0*3+c] + F[i*3+1]*V[1*3+c] + F[i*3+2]*V[2*3+c];
      atomicAdd(&sums[(size_t)r*9 + c*3 + i], lv);
    }
  }
  atomicAdd(&cnt[r], 1.0f);
}

// ---------------- kernel 3: scalar branch GEMM via V_WMMA_F32_16X16X4_F32 ----------------
// s = merged(16x160pad) @ Wp.T + b_so ; out_s = silu(s). Branch-free inner loop.
__global__ void k_scalar_out(const float* __restrict__ scalar, // (N,128)
                             const float* __restrict__ vnorm,  // (N,16)
                             const float* __restrict__ sums,   // (N,9)
                             const float* __restrict__ cnt,    // (N)
                             const float* __restrict__ Wp,     // (128,160) zero-padded
                             const float* __restrict__ b_so,   // (128)
                             float* __restrict__ out_s,        // (N,128)
                             int N){
  __shared__ float A[16][MERGED_PAD+1];   // stride 161 -> conflict-free column reads
  int nb = blockIdx.x*16;
  for (int idx = threadIdx.x; idx < 16*MERGED_PAD; idx += 256){
    int ln = idx / MERGED_PAD, col = idx % MERGED_PAD;
    int n = nb + ln;
    float v = 0.0f;
    if (n < N){
      if (col < S_IN)            v = scalar[(size_t)n*S_IN + col];
      else if (col < S_IN+HID)   v = vnorm[(size_t)n*HID + (col - S_IN)];
      else if (col < MERGED)     v = sums[(size_t)n*9 + (col - S_IN - HID)] / fmaxf(cnt[n], 1.0f);
    }
    A[ln][col] = v;
  }
  __syncthreads();
  int wave = threadIdx.x >> 5;   // 8 waves -> 8 column tiles of 16 (S_OUT=128)
  int lane = threadIdx.x & 31;
  int sub  = lane & 15;
  int half = lane >> 4;
  int ocol = wave*16 + sub;      // output column this lane serves in B/D
  const float* brow = &Wp[(size_t)ocol*MERGED_PAD];
  v8f acc = {0.f,0.f,0.f,0.f,0.f,0.f,0.f,0.f};
  #pragma unroll 4
  for (int k0 = 0; k0 < MERGED_PAD; k0 += 4){
    int kk = k0 + 2*half;
    v2f a, b;
    a.x = A[sub][kk];
    a.y = A[sub][kk+1];
    b.x = brow[kk];
    b.y = brow[kk+1];
    acc = __builtin_amdgcn_wmma_f32_16x16x4_f32(false, a, false, b, (short)0, acc, false, false);
  }
  float bias = b_so[ocol];
  float* orow = &out_s[(size_t)(nb + 8*half)*S_OUT + ocol];
  if (nb + 16 <= N){
    // full tile: unguarded epilogue (always taken when N % 16 == 0)
    #pragma unroll
    for (int r=0;r<8;r++){
      float s = acc[r] + bias;
      orow[(size_t)r*S_OUT] = s * sigmoidf_(s);
    }
  } else {
    #pragma unroll
    for (int r=0;r<8;r++){
      int n = nb + r + 8*half;   // C/D layout: reg r = row r (lanes 0-15) / r+8 (16-31)
      if (n < N){
        float s = acc[r] + bias;
        out_s[(size_t)n*S_OUT + ocol] = s * sigmoidf_(s);
      }
    }
  }
}

// ---------------- kernel 4: gate GEMM (WMMA) + vector output ----------------
// gate = silu_s(16x128) @ W_g.T(128x16) + b_g ; vrep = (vh @ W_up.T)^T * sigmoid(gate)
__global__ void k_vec_out(const float* __restrict__ silu_s, // (N,128) = first half of d_out
                          const float* __restrict__ vh,     // (N,3,16)
                          const float* __restrict__ W_g,    // (16,128)
                          const float* __restrict__ b_g,    // (16)
                          const float* __restrict__ W_up,   // (16,16)
                          float* __restrict__ out_v,        // (N,16,3)
                          int N){
  __shared__ float vhl[128*48];
  __shared__ float wup[V_OUT*HID];
  int nb = blockIdx.x*128;
  for (int idx = threadIdx.x; idx < 128*48; idx += 256){
    int n = nb + idx/48;
    vhl[idx] = (n < N) ? vh[(size_t)n*48 + idx%48] : 0.0f;
  }
  wup[threadIdx.x] = W_up[threadIdx.x];   // blockDim==256==16*16
  __syncthreads();
  int wave = threadIdx.x >> 5;            // each wave: one 16-node tile
  int lane = threadIdx.x & 31;
  int sub  = lane & 15;
  int half = lane >> 4;
  int tb = nb + wave*16;
  // clamp A row index instead of masking: dummy rows read node N-1; their D rows
  // are never stored, and valid rows are unaffected (D row depends only on its A row)
  int an = min(tb + sub, N-1);
  const float* arow = &silu_s[(size_t)an*S_OUT];
  const float* brow = &W_g[sub*S_OUT];
  v8f acc = {0.f,0.f,0.f,0.f,0.f,0.f,0.f,0.f};
  #pragma unroll 4
  for (int k0=0;k0<S_OUT;k0+=4){
    int kk = k0 + 2*half;
    v2f a, b;
    a.x = arow[kk];
    a.y = arow[kk+1];
    b.x = brow[kk];
    b.y = brow[kk+1];
    acc = __builtin_amdgcn_wmma_f32_16x16x4_f32(false, a, false, b, (short)0, acc, false, false);
  }
  int v = sub;
  float bias = b_g[v];
  bool full = (tb + 16 <= N);
  #pragma unroll
  for (int r=0;r<8;r++){
    int ln = wave*16 + r + 8*half;
    int n = nb + ln;
    if (full || n < N){
      float g = sigmoidf_(acc[r] + bias);
      const float* vrow = &vhl[ln*48];
      #pragma unroll
      for (int x=0;x<3;x++){
        float d = 0.f;
        #pragma unroll
        for (int hh=0;hh<HID;hh++) d += vrow[x*16+hh]*wup[v*16+hh];
        out_v[(size_t)n*48 + v*3 + x] = g*d;
      }
    }
  }
}

extern "C" void kernel_launch(void* const* d_in, const int* in_sizes, int n_in,
                              void* d_out, int out_size, void* d_ws, size_t ws_size,
                              hipStream_t stream) {
  const float* scalar = (const float*)d_in[0];
  const float* vector = (const float*)d_in[1];
  const int*   ei     = (const int*)  d_in[2];
  const float* frames = (const float*)d_in[3];
  const float* W_down = (const float*)d_in[4];
  const float* W_df   = (const float*)d_in[5];
  const float* W_so   = (const float*)d_in[6];
  const float* b_so   = (const float*)d_in[7];
  const float* W_up   = (const float*)d_in[8];
  const float* W_g    = (const float*)d_in[9];
  const float* b_g    = (const float*)d_in[10];

  int N = in_sizes[0] / S_IN;
  int E = in_sizes[2] / 2;

  // workspace layout (floats): vh | vnorm | vdf | sums | cnt | Wp
  float* ws    = (float*)d_ws;
  float* vh    = ws;                         // N*48
  float* vnorm = ws + (size_t)N*48;          // N*16
  float* vdf   = ws + (size_t)N*64;          // N*9
  float* sums  = ws + (size_t)N*73;          // N*9
  float* cnt   = ws + (size_t)N*82;          // N   (sums+cnt contiguous: N*10)
  float* Wp    = ws + (size_t)N*83;          // 128*160 zero-padded W_so

  float* out_s = (float*)d_out;              // (N,128) silu(s)
  float* out_v = out_s + (size_t)N*S_OUT;    // (N,16,3) vrep

  int nzero = N*10;
  k_zero<<<(nzero+255)/256, 256, 0, stream>>>(sums, nzero);
  k_pad_wso<<<(S_OUT*MERGED_PAD)/256, 256, 0, stream>>>(W_so, Wp);
  k_node_prep<<<(N+15)/16, 256, 0, stream>>>(vector, W_down, W_df, vh, vnorm, vdf, N);
  k_edge_scatter<<<(E+255)/256, 256, 0, stream>>>(ei, frames, vdf, sums, cnt, E);
  k_scalar_out<<<(N+15)/16, 256, 0, stream>>>(scalar, vnorm, sums, cnt, Wp, b_so, out_s, N);
  k_vec_out<<<(N+127)/128, 256, 0, stream>>>(out_s, vh, W_g, b_g, W_up, out_v, N);
}